// EHRGNN_motor_20667382628838
// MI455X (gfx1250) — compile-verified
//
#include <hip/hip_runtime.h>

// ---------------------------------------------------------------------------
// 2-layer GCN on gfx1250 (MI455X).
//   GEMMs        : V_WMMA_F32_16X16X4_F32, LDS-staged A tiles, prefetch
//   edge scatter : wave32-per-edge streaming + global_atomic_add_f32
// ---------------------------------------------------------------------------

typedef __attribute__((ext_vector_type(2))) float v2f;
typedef __attribute__((ext_vector_type(8))) float v8f;

#define CH 128          // IN_CH == HID_CH == OUT_CH
#define KC 64           // K-chunk staged in LDS
#define LDS_STRIDE 68   // 16 rows x (64+4) floats: pad to de-conflict banks

// ------------------------ GEMM: C[Mx128] = A[MxK] @ W[Kx128] (+bias) -------
// grid.x = M/16, block = 256 threads (8 waves). Wave w owns N-tile w.
__global__ void gemm_wmma_f32(const float* __restrict__ A,
                              const float* __restrict__ W,
                              const float* __restrict__ bias,
                              float* __restrict__ C,
                              int K)
{
    __shared__ float As[16 * LDS_STRIDE];

    const int t    = threadIdx.x;
    const int lane = t & 31;
    const int wave = t >> 5;
    const int half = lane >> 4;   // 0: lanes 0-15, 1: lanes 16-31
    const int rc   = lane & 15;   // row (A) / col (B,C) within tile
    const int n0   = wave * 16;   // this wave's N-tile base column
    const int m0   = blockIdx.x * 16;

    // cooperative A-chunk load: 256 threads x float4 == 16 x 64 floats
    const int lr = t >> 4;              // 0..15  (row)
    const int lc = (t & 15) << 2;       // 0..60  (col, float4)

    const float* Wp = W + n0 + rc;      // B-fragment column base

    v8f acc = {};
    for (int kc = 0; kc < K; kc += KC) {
        __syncthreads();
        const float4 a4 =
            *(const float4*)(A + (size_t)(m0 + lr) * K + kc + lc);
        As[lr * LDS_STRIDE + lc + 0] = a4.x;
        As[lr * LDS_STRIDE + lc + 1] = a4.y;
        As[lr * LDS_STRIDE + lc + 2] = a4.z;
        As[lr * LDS_STRIDE + lc + 3] = a4.w;
        __syncthreads();

        if (kc + KC < K)  // emits global_prefetch_b8 for the next A chunk
            __builtin_prefetch(A + (size_t)(m0 + lr) * K + kc + KC + lc, 0, 0);

#pragma unroll
        for (int kk = 0; kk < KC; kk += 4) {
            const int ko = kk + 2 * half;
            v2f a, b;
            a.x = As[rc * LDS_STRIDE + ko];       // A[m=rc][k0+2h]
            a.y = As[rc * LDS_STRIDE + ko + 1];   // A[m=rc][k0+2h+1]
            b.x = Wp[(size_t)(kc + ko) * CH];     // W[k0+2h][n0+rc]
            b.y = Wp[(size_t)(kc + ko + 1) * CH]; // W[k0+2h+1][n0+rc]
            acc = __builtin_amdgcn_wmma_f32_16x16x4_f32(
                false, a, false, b, (short)0, acc, false, false);
        }
    }

    const int col = n0 + rc;
    const float bv = bias ? bias[col] : 0.0f;
#pragma unroll
    for (int i = 0; i < 8; ++i) {                 // D vgpr i -> row i + 8*half
        const int m = m0 + i + 8 * half;
        C[(size_t)m * CH + col] = acc[i] + bv;
    }
}

// ------------------------ degree / norm ------------------------------------
__global__ void deg_init_k(float* __restrict__ deg, int n) {
    int i = blockIdx.x * blockDim.x + threadIdx.x;
    if (i < n) deg[i] = 1.0f;                     // self-loop pre-counted
}
__global__ void deg_accum_k(const int* __restrict__ dst,
                            float* __restrict__ deg, int E) {
    int e = blockIdx.x * blockDim.x + threadIdx.x;
    if (e < E) atomicAdd(&deg[dst[e]], 1.0f);
}
__global__ void dinv_k(const float* __restrict__ deg,
                       float* __restrict__ dinv, int n) {
    int i = blockIdx.x * blockDim.x + threadIdx.x;
    if (i < n) dinv[i] = __frsqrt_rn(deg[i]);     // deg >= 1 always
}

// ------------------------ elementwise helpers ------------------------------
__global__ void copy4_k(const float4* __restrict__ in,
                        float4* __restrict__ out, int n4) {
    int i = blockIdx.x * blockDim.x + threadIdx.x;
    if (i < n4) out[i] = in[i];
}
// acc[i][c] = bias[c] + h[i][c] * dinv[i]^2   (self-loop + bias folded in)
__global__ void init_acc_k(const float* __restrict__ h,
                           const float* __restrict__ dinv,
                           const float* __restrict__ bias,
                           float* __restrict__ acc, int n) {
    int g = blockIdx.x * blockDim.x + threadIdx.x;
    if (g < n) {
        const int   i  = g >> 7;          // /128
        const int   c  = g & (CH - 1);
        const float di = dinv[i];
        acc[g] = bias[c] + h[g] * di * di;
    }
}
__global__ void relu_k(const float* __restrict__ in,
                       float* __restrict__ out, int n) {
    int g = blockIdx.x * blockDim.x + threadIdx.x;
    if (g < n) out[g] = fmaxf(in[g], 0.0f);
}

// ------------------------ edge scatter: wave32 per edge --------------------
// acc[dst] += h[src] * dinv[src]*dinv[dst]; lane l handles channels 4l..4l+3
__global__ void edge_scatter_k(const int* __restrict__ src,
                               const int* __restrict__ dst,
                               const float* __restrict__ dinv,
                               const float* __restrict__ h,
                               float* __restrict__ acc, int E) {
    const int wid  = (blockIdx.x * blockDim.x + threadIdx.x) >> 5;
    const int lane = threadIdx.x & 31;
    if (wid >= E) return;
    const int   s = src[wid];
    const int   d = dst[wid];
    const float w = dinv[s] * dinv[d];
    const float4 v = ((const float4*)(h + (size_t)s * CH))[lane];
    float* o = acc + (size_t)d * CH + lane * 4;
    atomicAdd(o + 0, v.x * w);
    atomicAdd(o + 1, v.y * w);
    atomicAdd(o + 2, v.z * w);
    atomicAdd(o + 3, v.w * w);
}

// ---------------------------------------------------------------------------
extern "C" void kernel_launch(void* const* d_in, const int* in_sizes, int n_in,
                              void* d_out, int out_size, void* d_ws, size_t ws_size,
                              hipStream_t stream) {
    const float* init_emb = (const float*)d_in[0];
    const int*   eidx     = (const int*)  d_in[1];
    const float* map_W    = (const float*)d_in[2];
    const float* map_b    = (const float*)d_in[3];
    const float* s_emb    = (const float*)d_in[4];
    const float* W1       = (const float*)d_in[5];
    const float* b1       = (const float*)d_in[6];
    const float* W2       = (const float*)d_in[7];
    const float* b2       = (const float*)d_in[8];
    float*       out      = (float*)d_out;

    const int E     = in_sizes[1] / 2;        // 1,600,000
    const int Kmap  = in_sizes[2] / CH;       // 768
    const int NEID  = in_sizes[0] / Kmap;     // 20,000
    const int NS    = in_sizes[4] / CH;       // 80,000
    const int N     = NEID + NS;              // 100,000
    const int NT    = N * CH;                 // 12.8M elems

    // workspace: x0 | h | acc | deg | dinv   (~155 MB)
    float* x0   = (float*)d_ws;
    float* h    = x0  + (size_t)NT;
    float* acc  = h   + (size_t)NT;
    float* deg  = acc + (size_t)NT;
    float* dinv = deg + (size_t)N;

    const int* src = eidx;
    const int* dst = eidx + E;

    const int TB = 256;
    // degrees / symmetric norm
    deg_init_k <<<(N + TB - 1) / TB, TB, 0, stream>>>(deg, N);
    deg_accum_k<<<(E + TB - 1) / TB, TB, 0, stream>>>(dst, deg, E);
    dinv_k     <<<(N + TB - 1) / TB, TB, 0, stream>>>(deg, dinv, N);

    // eid embedding map + concat:  x0 = [init_emb@map_W + map_b ; s_emb]
    gemm_wmma_f32<<<NEID / 16, TB, 0, stream>>>(init_emb, map_W, map_b, x0, Kmap);
    copy4_k<<<((NS * (CH / 4)) + TB - 1) / TB, TB, 0, stream>>>(
        (const float4*)s_emb, (float4*)(x0 + (size_t)NEID * CH), NS * (CH / 4));

    // ---- conv1 ----
    gemm_wmma_f32<<<N / 16, TB, 0, stream>>>(x0, W1, nullptr, h, CH);
    init_acc_k   <<<(NT + TB - 1) / TB, TB, 0, stream>>>(h, dinv, b1, acc, NT);
    edge_scatter_k<<<(E + 7) / 8, TB, 0, stream>>>(src, dst, dinv, h, acc, E);
    relu_k       <<<(NT + TB - 1) / TB, TB, 0, stream>>>(acc, x0, NT);

    // ---- conv2 (accumulate straight into d_out) ----
    gemm_wmma_f32<<<N / 16, TB, 0, stream>>>(x0, W2, nullptr, h, CH);
    init_acc_k   <<<(NT + TB - 1) / TB, TB, 0, stream>>>(h, dinv, b2, out, NT);
    edge_scatter_k<<<(E + 7) / 8, TB, 0, stream>>>(src, dst, dinv, h, out, E);
}